// vp_layer_conjgrad_11922829213826
// MI455X (gfx1250) — compile-verified
//
#include <hip/hip_runtime.h>
#include <math.h>

// ---------------------------------------------------------------------------
// vp_layer_conjgrad on MI455X (gfx1250):
//   fft2/ifft2 as complex 512^3 GEMMs on V_WMMA_F32_16X16X4_F32.
//   ALL GEMM operand traffic moves via the Tensor Data Mover into LDS:
//     - A strip (16x512 re+im) loaded once, bank-conflict-free via TDM padding
//     - B staged in 64x128 double-buffered chunks, prefetch overlapped with
//       WMMA using in-order TENSORcnt semantics (s_wait_tensorcnt N)
//   GEMM epilogue variants are compile-time template flags (no runtime
//   branches in the K loop).
// ---------------------------------------------------------------------------

typedef float v2f  __attribute__((ext_vector_type(2)));
typedef float v8f  __attribute__((ext_vector_type(8)));
typedef unsigned int u32x4 __attribute__((ext_vector_type(4)));
typedef int  i32x4 __attribute__((ext_vector_type(4)));
typedef int  i32x8 __attribute__((ext_vector_type(8)));

#define N512   512
#define PLANE  262144        // 512*512
#define NIMG   6             // 2*3
#define NTOT   (NIMG*PLANE)  // 1572864
#define LAMBDA 0.025f

// GEMM template flags
#define FB_BCOMPLEX 1   // B has imaginary part
#define FB_WRITECI  2   // store imaginary output plane
#define FB_ABS      4   // store |re + i*im| into Cr
#define FB_ASTRIDE  8   // A is per-image (stride PLANE by blockIdx.z)
#define FB_BSTRIDE 16   // B is per-image
#define FB_ADDLAM  32   // Cr += LAMBDA * Padd

// LDS layout (float indices); A rows padded to stride 516 by the TDM
#define LDS_ARE   0
#define LDS_AIM   8320          // 16*516 = 8256, rounded up
#define LDS_B     16640
#define LDS_BUFSZ 16384         // one B buffer: re(8192) + im(8192)
#define LDS_BYTES 197632        // 49408 floats

// TDM pad: every 256 dwords insert 2 dwords -> A row stride 516 floats
#define TDM_PAD_BITS ((1u << 20) | (7u << 22) | (1u << 25))

__device__ inline v8f wmma_f32x4(v2f a, v2f b, v8f c) {
  return __builtin_amdgcn_wmma_f32_16x16x4_f32(false, a, false, b, (short)0, c,
                                               false, false);
}

// TDM: 2D tile (tileH rows x tileW f32, row stride 512 f32) -> LDS.
__device__ inline void tdm_load_2d(const float* gptr, unsigned ldsByteOff,
                                   int tileW, int tileH, unsigned w0extra) {
  unsigned long long ga = (unsigned long long)(uintptr_t)gptr;
  u32x4 g0 = { 1u,                                   // count=1 valid descriptor
               ldsByteOff,
               (unsigned)(ga & 0xFFFFFFFFu),
               ((unsigned)((ga >> 32) & 0x01FFFFFFu)) | 0x80000000u }; // type=2
  i32x8 g1 = { (int)(0x00020000u | w0extra),  // data_size=4B (+pad ctl)
               (int)(N512 << 16),             // tensor_dim0 = 512
               (int)(N512 << 16),             // tensor_dim1 = 512
               (int)(tileW << 16),            // tile_dim0
               tileH,                         // tile_dim1
               N512,                          // tensor_dim0_stride = 512
               0, 0 };
  i32x4 z4 = {0, 0, 0, 0};
#if __clang_major__ >= 23
  i32x8 z8 = {0, 0, 0, 0, 0, 0, 0, 0};
  __builtin_amdgcn_tensor_load_to_lds(g0, g1, z4, z4, z8, 0);
#else
  __builtin_amdgcn_tensor_load_to_lds(g0, g1, z4, z4, 0);
#endif
}

// ---------------------------------------------------------------------------
// Complex GEMM: C(512x512) = A * B, one 16x16 complex tile per wave.
// Block = 256 threads = 8 waves -> M=16, N=128 per workgroup.
// grid = (4, 32, 6). K chunked at 64; B chunks double-buffered via TDM.
// ---------------------------------------------------------------------------
template <int FLAGS>
__global__ __launch_bounds__(256)
void k_cgemm(float* __restrict__ Cr, float* __restrict__ Ci,
             const float* __restrict__ Ar, const float* __restrict__ Ai,
             const float* __restrict__ Br, const float* __restrict__ Bi,
             const float* __restrict__ Padd) {
  constexpr bool BCX   = (FLAGS & FB_BCOMPLEX) != 0;
  constexpr bool WCI   = (FLAGS & FB_WRITECI) != 0;
  constexpr bool ABSO  = (FLAGS & FB_ABS) != 0;
  constexpr bool ADDL  = (FLAGS & FB_ADDLAM) != 0;
  constexpr bool NEEDI = WCI || ABSO;
  constexpr int  TPC   = BCX ? 2 : 1;       // TDM issues per B chunk

  extern __shared__ float lds[];

  const int z    = blockIdx.z;
  const int aoff = (FLAGS & FB_ASTRIDE) ? z * PLANE : 0;
  const int boff = (FLAGS & FB_BSTRIDE) ? z * PLANE : 0;
  const int m0   = blockIdx.y << 4;
  const int n0   = blockIdx.x << 7;

  const int lane = threadIdx.x & 31;
  const int wave = threadIdx.x >> 5;
  const int row  = lane & 15;
  const int hi   = lane >> 4;               // K sub-pair select (ISA A layout)
  const int colL = (wave << 4) + row;       // 0..127 inside WG N-block
  const int col  = n0 + colL;

  const float* Bre = Br + boff + n0;
  const float* Bim = Bi + boff + n0;        // unused unless BCX

  // Prologue: A strip (padded) + B chunk 0 into buffer 0.
  if (threadIdx.x < 32) {
    unsigned base = (unsigned)__builtin_amdgcn_groupstaticsize();
    tdm_load_2d(Ar + aoff + m0 * N512, base + LDS_ARE * 4u, 512, 16, TDM_PAD_BITS);
    tdm_load_2d(Ai + aoff + m0 * N512, base + LDS_AIM * 4u, 512, 16, TDM_PAD_BITS);
    tdm_load_2d(Bre, base + LDS_B * 4u, 128, 64, 0);
    if constexpr (BCX) tdm_load_2d(Bim, base + (LDS_B + 8192) * 4u, 128, 64, 0);
  }

  v8f accr = {0.f, 0.f, 0.f, 0.f, 0.f, 0.f, 0.f, 0.f};
  v8f acci = {0.f, 0.f, 0.f, 0.f, 0.f, 0.f, 0.f, 0.f};
  const int lA = row * 516;                 // padded A row stride

  for (int c = 0; c < 8; ++c) {
    __syncthreads();   // compute using the buffer we are about to overwrite is done
    if (threadIdx.x < 32) {
      if (c + 1 < 8) {
        unsigned base = (unsigned)__builtin_amdgcn_groupstaticsize();
        unsigned dst  = base + (LDS_B + ((c + 1) & 1) * LDS_BUFSZ) * 4u;
        tdm_load_2d(Bre + (c + 1) * 64 * N512, dst, 128, 64, 0);
        if constexpr (BCX)
          tdm_load_2d(Bim + (c + 1) * 64 * N512, dst + 8192u * 4u, 128, 64, 0);
        // TDM completes in order: <=TPC outstanding => A + chunk c resident.
        __builtin_amdgcn_s_wait_tensorcnt(TPC);
      } else {
        __builtin_amdgcn_s_wait_tensorcnt(0);
      }
    }
    __syncthreads();   // chunk c visible to all waves
    const int lB   = LDS_B + (c & 1) * LDS_BUFSZ;
    const int pad2 = (c >= 4) ? 2 : 0;      // TDM pad offset for k >= 256
    const int aBase = lA + (c << 6) + pad2;

#pragma unroll 4
    for (int k = 0; k < 64; k += 4) {
      const int kk = k + (hi << 1);
      v2f ar = *(const v2f*)&lds[aBase + kk];
      v2f ai = *(const v2f*)&lds[LDS_AIM + aBase + kk];
      v2f br = { lds[lB + kk * 128 + colL], lds[lB + (kk + 1) * 128 + colL] };

      accr = wmma_f32x4(ar, br, accr);
      if constexpr (NEEDI) acci = wmma_f32x4(ai, br, acci);
      if constexpr (BCX) {
        v2f bi  = { lds[lB + 8192 + kk * 128 + colL],
                    lds[lB + 8192 + (kk + 1) * 128 + colL] };
        v2f nai = { -ai.x, -ai.y };         // f32 WMMA has no A/B NEG bits
        accr = wmma_f32x4(nai, bi, accr);
        if constexpr (NEEDI) acci = wmma_f32x4(ar, bi, acci);
      }
    }
  }

  float* Crp = Cr + z * PLANE;
  float* Cip = Ci + z * PLANE;
  const float* Pp = Padd + z * PLANE;

#pragma unroll
  for (int v = 0; v < 8; ++v) {
    const int rrow = m0 + v + (hi << 3);    // C layout: vgpr v -> M=v (+8 hi half)
    const int idx  = rrow * N512 + col;
    float re = accr[v];
    if constexpr (ADDL) re += LAMBDA * Pp[idx];
    if constexpr (ABSO) re = sqrtf(re * re + acci[v] * acci[v]);
    Crp[idx] = re;
    if constexpr (WCI) Cip[idx] = acci[v];
  }
}

// ---------------------------------------------------------------------------
// Pointwise / setup kernels
// ---------------------------------------------------------------------------
__global__ void k_init_dft(float* Fr, float* Fi, float* Gr, float* Gi) {
  int i = blockIdx.x * 256 + threadIdx.x;        // PLANE threads
  int rr = i >> 9, cc = i & 511;
  int t = (rr * cc) & 511;
  float th = 6.283185307179586f * (float)t * (1.0f / 512.0f);
  float sn, cs;
  __sincosf(th, &sn, &cs);
  Fr[i] = cs;  Fi[i] = -sn;                      // e^{-i th}
  Gr[i] = cs * (1.0f / 512.0f);                  // conj(F)/N
  Gi[i] = sn * (1.0f / 512.0f);
}

__global__ void k_ifftshift(float* dst, const float* src) {
  int i = blockIdx.x * 256 + threadIdx.x;        // NTOT threads
  int z = i >> 18, rem = i & (PLANE - 1);
  int rr = rem >> 9, cc = rem & 511;
  dst[i] = src[(z << 18) + (((rr + 256) & 511) << 9) + ((cc + 256) & 511)];
}

__global__ void k_wiener(const float* Yr, const float* Yi,
                         const float* Kr, const float* Ki,
                         float* D, float* xfR, float* xfI,
                         float* resR, float* resI) {
  int i = blockIdx.x * 256 + threadIdx.x;        // NTOT threads
  float kr = Kr[i], ki = Ki[i], yr = Yr[i], yi = Yi[i];
  float d  = kr * kr + ki * ki;
  float dd = d + LAMBDA;
  float br = yr * kr + yi * ki;                  // b = Y * conj(K)
  float bi = yi * kr - yr * ki;
  float xr = br / dd, xi = bi / dd;
  D[i] = d; xfR[i] = xr; xfI[i] = xi;
  resR[i] = br - xr * dd;
  resI[i] = bi - xi * dd;
}

__global__ void k_mulkk(float* Wr, float* Wi, const float* D) {
  int i = blockIdx.x * 256 + threadIdx.x;
  float d = D[i];
  Wr[i] *= d; Wi[i] *= d;
}

__global__ void k_update_p(float* p, const float* r, const float* scal, int it) {
  int i = blockIdx.x * 256 + threadIdx.x;
  float beta = scal[2];
  p[i] = (it > 1) ? (r[i] + beta * p[i]) : r[i];
}

__global__ void k_update_xr(float* x, float* r, const float* p, const float* Ap,
                            const float* scal) {
  int i = blockIdx.x * 256 + threadIdx.x;
  float a = scal[3];
  x[i] += a * p[i];
  r[i] -= a * Ap[i];
}

__global__ void k_scal_init(float* scal) {
  scal[0] = 0.f; scal[1] = 1.f; scal[2] = 0.f;   // rho, rho_1, beta
  scal[3] = 0.f; scal[4] = 0.f; scal[5] = 0.f;   // alpha, done, pap
}

__global__ __launch_bounds__(256)
void k_reduce_dot(const float* a, const float* b, float* partial, int n) {
  __shared__ float sm[256];
  float s = 0.f;
  for (int i = blockIdx.x * blockDim.x + threadIdx.x; i < n;
       i += gridDim.x * blockDim.x)
    s += a[i] * b[i];
  sm[threadIdx.x] = s;
  __syncthreads();
  for (int o = 128; o > 0; o >>= 1) {
    if (threadIdx.x < o) sm[threadIdx.x] += sm[threadIdx.x + o];
    __syncthreads();
  }
  if (threadIdx.x == 0) partial[blockIdx.x] = sm[0];
}

__global__ __launch_bounds__(256)
void k_finalize(const float* partial, float* scal, int mode) {
  __shared__ float sm[256];
  float s = 0.f;
  for (int i = threadIdx.x; i < 1024; i += 256) s += partial[i];
  sm[threadIdx.x] = s;
  __syncthreads();
  for (int o = 128; o > 0; o >>= 1) {
    if (threadIdx.x < o) sm[threadIdx.x] += sm[threadIdx.x + o];
    __syncthreads();
  }
  if (threadIdx.x == 0) {
    float tot = sm[0];
    if (mode == 0) {                 // rho path
      float rho = tot;
      scal[2] = rho / scal[1];       // beta = rho / rho_1
      scal[4] = (rho < 1e-15f) ? 1.f : 0.f;   // done (freeze updates)
      scal[0] = rho;
      scal[1] = rho;
    } else {                         // alpha path
      scal[5] = tot;
      scal[3] = (scal[4] != 0.f) ? 0.f : (scal[0] / tot);
    }
  }
}

__global__ void k_crop(float* out, const float* x) {
  int i = blockIdx.x * 256 + threadIdx.x;
  if (i >= NIMG * 480 * 480) return;
  int z = i / (480 * 480);
  int rem = i % (480 * 480);
  int rr = rem / 480, cc = rem % 480;
  out[i] = x[z * PLANE + (rr + 16) * N512 + (cc + 16)];
}

// ---------------------------------------------------------------------------
// Host orchestration (graph-capture safe: stream-only, no alloc/sync)
// ---------------------------------------------------------------------------
extern "C" void kernel_launch(void* const* d_in, const int* in_sizes, int n_in,
                              void* d_out, int out_size, void* d_ws, size_t ws_size,
                              hipStream_t stream) {
  (void)in_sizes; (void)n_in; (void)out_size; (void)ws_size;
  const float* y   = (const float*)d_in[0];
  const float* kin = (const float*)d_in[1];
  float* out = (float*)d_out;

  float* W = (float*)d_ws;
  const size_t P = PLANE;
  float* Fr = W;            float* Fi = W + P;
  float* Gr = W + 2 * P;    float* Gi = W + 3 * P;
  float* kk = W + 4 * P;                        // |K|^2, 6 planes
  float* x  = W + 10 * P;                       // 6 planes each
  float* r  = W + 16 * P;
  float* p  = W + 22 * P;
  float* Ap = W + 28 * P;
  float* S0 = W + 34 * P;                       // scratch, type-major 6-plane blocks
  float* S1 = S0 + 6 * P;  float* S2 = S0 + 12 * P; float* S3 = S0 + 18 * P;
  float* S4 = S0 + 24 * P; float* S5 = S0 + 30 * P;
  float* S6 = S0 + 36 * P; float* S7 = S0 + 42 * P;
  float* partials = W + 82 * P;                 // 1024 floats
  float* scal     = partials + 2048;

  const dim3 gG(4, 32, NIMG);
#define GEMM(FL, Crr, Cii, Arr, Aii, Brr, Bii, Pa)                          \
  k_cgemm<(FL)><<<gG, 256, LDS_BYTES, stream>>>(Crr, Cii, Arr, Aii, Brr, Bii, Pa)
  const int EW = NTOT / 256;   // 6144 blocks for elementwise kernels

  // ---- setup: DFT matrices, ifftshift(k) -------------------------------
  k_init_dft<<<PLANE / 256, 256, 0, stream>>>(Fr, Fi, Gr, Gi);
  k_ifftshift<<<EW, 256, 0, stream>>>(S4, kin);

  // ---- fft2(y): T = F*y ; Y = T*F --------------------------------------
  GEMM(FB_BSTRIDE | FB_WRITECI, S0, S1, Fr, Fi, y, Fr, S0);
  GEMM(FB_ASTRIDE | FB_BCOMPLEX | FB_WRITECI, S2, S3, S0, S1, Fr, Fi, S2);
  // ---- fft2(ifftshift(k)) ----------------------------------------------
  GEMM(FB_BSTRIDE | FB_WRITECI, S6, S7, Fr, Fi, S4, Fr, S6);
  GEMM(FB_ASTRIDE | FB_BCOMPLEX | FB_WRITECI, S4, S5, S6, S7, Fr, Fi, S4);
  // ---- Wiener init: kk, x_fft, residual --------------------------------
  k_wiener<<<EW, 256, 0, stream>>>(S2, S3, S4, S5, kk, S0, S1, S6, S7);
  // ---- x0 = |ifft2(x_fft)| ---------------------------------------------
  GEMM(FB_BSTRIDE | FB_BCOMPLEX | FB_WRITECI, S2, S3, Gr, Gi, S0, S1, S2);
  GEMM(FB_ASTRIDE | FB_BCOMPLEX | FB_ABS, x, x, S2, S3, Gr, Gi, x);
  // ---- r0 = |ifft2(b - Ax)| --------------------------------------------
  GEMM(FB_BSTRIDE | FB_BCOMPLEX | FB_WRITECI, S2, S3, Gr, Gi, S6, S7, S2);
  GEMM(FB_ASTRIDE | FB_BCOMPLEX | FB_ABS, r, r, S2, S3, Gr, Gi, r);
  k_scal_init<<<1, 1, 0, stream>>>(scal);

  // ---- 50 CG iterations -------------------------------------------------
  for (int it = 0; it < 50; ++it) {
    k_reduce_dot<<<1024, 256, 0, stream>>>(r, r, partials, NTOT);
    k_finalize<<<1, 256, 0, stream>>>(partials, scal, 0);
    k_update_p<<<EW, 256, 0, stream>>>(p, r, scal, it);
    // W = fft2(p)
    GEMM(FB_BSTRIDE | FB_WRITECI, S0, S1, Fr, Fi, p, Fr, S0);
    GEMM(FB_ASTRIDE | FB_BCOMPLEX | FB_WRITECI, S2, S3, S0, S1, Fr, Fi, S2);
    // W *= |K|^2
    k_mulkk<<<EW, 256, 0, stream>>>(S2, S3, kk);
    // Ap = real(ifft2(W)) + lambda*p
    GEMM(FB_BSTRIDE | FB_BCOMPLEX | FB_WRITECI, S0, S1, Gr, Gi, S2, S3, S0);
    GEMM(FB_ASTRIDE | FB_BCOMPLEX | FB_ADDLAM, Ap, Ap, S0, S1, Gr, Gi, p);
    k_reduce_dot<<<1024, 256, 0, stream>>>(p, Ap, partials, NTOT);
    k_finalize<<<1, 256, 0, stream>>>(partials, scal, 1);
    k_update_xr<<<EW, 256, 0, stream>>>(x, r, p, Ap, scal);
  }

  // ---- center crop to (480,480) ----------------------------------------
  k_crop<<<(NIMG * 480 * 480 + 255) / 256, 256, 0, stream>>>(out, x);
#undef GEMM
}